// GCN_Processor_9294309228961
// MI455X (gfx1250) — compile-verified
//
#include <hip/hip_runtime.h>

// GCN (PyG-default gcn_norm) for MI455X / gfx1250, wave32.
// Heavy phase: per-edge scatter-add (L2-resident, hardware f32 atomics).
// GEMM phase: V_WMMA_F32_16X16X4_F32 (fp32 WMMA), W staged in LDS with
// K-pair-interleaved layout so each B operand is one aligned ds_load_b64.

typedef __attribute__((ext_vector_type(2))) float v2f;
typedef __attribute__((ext_vector_type(8))) float v8f;

#define DHID 128
#define KPSTRIDE 288   // floats per k-pair row in LDS (256 data + 32 pad => +32-bank skew)

// ---------------------------------------------------------------- utilities
__global__ void k_fill1(float* __restrict__ p, int n) {
  int i = blockIdx.x * blockDim.x + threadIdx.x;
  if (i < n) p[i] = 1.0f;
}

__global__ void k_zero4(float4* __restrict__ p, int n4) {
  int i = blockIdx.x * blockDim.x + threadIdx.x;
  if (i < n4) p[i] = make_float4(0.f, 0.f, 0.f, 0.f);
}

__global__ void k_copy4(const float4* __restrict__ s, float4* __restrict__ d, int n4) {
  int i = blockIdx.x * blockDim.x + threadIdx.x;
  if (i < n4) d[i] = s[i];
}

// ------------------------------------------------------- degree / gcn_norm
__global__ void k_count_deg(const int* __restrict__ dst, float* __restrict__ deg, int E) {
  int e = blockIdx.x * blockDim.x + threadIdx.x;
  if (e < E) unsafeAtomicAdd(&deg[dst[e]], 1.0f);   // global_atomic_add_f32
}

__global__ void k_node_norm(float* __restrict__ deg, float* __restrict__ invdeg, int n) {
  int i = blockIdx.x * blockDim.x + threadIdx.x;
  if (i < n) {
    float d = deg[i];
    deg[i]    = rsqrtf(d);   // deg now holds deg^{-1/2}
    invdeg[i] = 1.0f / d;    // self-loop norm
  }
}

__global__ void k_edge_norm(const int* __restrict__ src, const int* __restrict__ dst,
                            const float* __restrict__ rs, float* __restrict__ ne, int E) {
  int e = blockIdx.x * blockDim.x + threadIdx.x;
  if (e < E) ne[e] = rs[src[e]] * rs[dst[e]];
}

// ---------------------------------------------------------------- fp32 WMMA GEMM
// y[N,128] = x[N,128] @ W[128,128].  One wave -> 16 rows x 128 cols.
// A 16x4 layout : lane m = lane&15, K-pair = (lane>>4)*2 in {v0,v1}
// B 4x16 layout : lane n = lane&15, K-pair = (lane>>4)*2 in {v0,v1}
// C/D 16x16     : vgpr i, row = i + 8*(lane>>4), col = lane&15
// LDS layout    : Wlds[(k>>1)*KPSTRIDE + n*2 + (k&1)] = W[k*128+n]
//                 -> B operand (W[k][n], W[k+1][n]) is one 8B ds_load_b64;
//                 KPSTRIDE=288 skews kp rows by 32 banks so lane halves
//                 (reading kp vs kp+1) never collide.
__global__ void __launch_bounds__(256)
k_gemm(const float* __restrict__ x, const float* __restrict__ W,
       float* __restrict__ y, int nrows) {
  __shared__ float Wlds[(DHID / 2) * KPSTRIDE];  // 72 KB of the 320 KB WGP LDS

  { // stage W interleaved: float4 global reads, scalar LDS scatter
    const float4* Wg = (const float4*)W;
    for (int i4 = threadIdx.x; i4 < DHID * DHID / 4; i4 += 256) {
      int k = i4 >> 5;                 // 32 float4 per 128-wide row
      int nb = (i4 & 31) << 2;
      float4 v = Wg[i4];
      float* wp = Wlds + (k >> 1) * KPSTRIDE + (k & 1);
      wp[(nb + 0) * 2] = v.x;
      wp[(nb + 1) * 2] = v.y;
      wp[(nb + 2) * 2] = v.z;
      wp[(nb + 3) * 2] = v.w;
    }
  }
  __syncthreads();

  const int wave = threadIdx.x >> 5;
  const int lane = threadIdx.x & 31;
  const int M0   = blockIdx.x * 128 + wave * 16;
  const int lm   = lane & 15;
  const int koff = (lane >> 4) << 1;

  int row  = M0 + lm;
  int rowc = row < nrows ? row : nrows - 1;      // clamp: garbage rows never stored
  const float* xp = x + (size_t)rowc * DHID;

  v8f acc[8] = {};

  for (int k = 0; k < DHID; k += 4) {
    v2f a = *(const v2f*)(xp + k + koff);                        // 8B global load
    const v2f* brow = (const v2f*)(Wlds + ((k + koff) >> 1) * KPSTRIDE);
#pragma unroll
    for (int t = 0; t < 8; ++t) {
      v2f b = brow[(t << 4) + lm];                               // one ds_load_b64
      acc[t] = __builtin_amdgcn_wmma_f32_16x16x4_f32(
          false, a, false, b, (short)0, acc[t], false, false);
    }
  }

  const int rbase = M0 + ((lane >> 4) << 3);
#pragma unroll
  for (int t = 0; t < 8; ++t) {
#pragma unroll
    for (int i = 0; i < 8; ++i) {
      int r = rbase + i;
      if (r < nrows) y[(size_t)r * DHID + (t << 4) + lm] = acc[t][i];
    }
  }
}

// ---------------------------------------------------------------- edge scatter
// one wave32 per edge: lane covers 4 channels (32*4 = 128)
__global__ void k_scatter(const float* __restrict__ y, const int* __restrict__ src,
                          const int* __restrict__ dst, const float* __restrict__ ne,
                          float* __restrict__ agg, int E) {
  int gid  = blockIdx.x * blockDim.x + threadIdx.x;
  int e    = gid >> 5;
  int lane = gid & 31;
  if (e >= E) return;
  int s = src[e], d = dst[e];
  float w = ne[e];
  float4 v = ((const float4*)(y + (size_t)s * DHID))[lane];
  float* ap = agg + (size_t)d * DHID + lane * 4;
  unsafeAtomicAdd(ap + 0, v.x * w);
  unsafeAtomicAdd(ap + 1, v.y * w);
  unsafeAtomicAdd(ap + 2, v.z * w);
  unsafeAtomicAdd(ap + 3, v.w * w);
}

// ---------------------------------------------------------------- finalize
// out = agg + y*inv_deg[row] + bias   (+ relu for inner layers)
__global__ void k_finalize(const float* __restrict__ agg, const float* __restrict__ y,
                           const float* __restrict__ invdeg, const float* __restrict__ bias,
                           float* __restrict__ out, int relu, int total4) {
  int i = blockIdx.x * blockDim.x + threadIdx.x;
  if (i >= total4) return;
  int row = i >> 5;               // 32 float4 per 128-wide row
  int c4  = (i & 31) * 4;
  float id = invdeg[row];
  float4 a  = ((const float4*)agg)[i];
  float4 yy = ((const float4*)y)[i];
  float4 bb = *(const float4*)(bias + c4);
  float4 r;
  r.x = a.x + yy.x * id + bb.x;
  r.y = a.y + yy.y * id + bb.y;
  r.z = a.z + yy.z * id + bb.z;
  r.w = a.w + yy.w * id + bb.w;
  if (relu) {
    r.x = fmaxf(r.x, 0.f); r.y = fmaxf(r.y, 0.f);
    r.z = fmaxf(r.z, 0.f); r.w = fmaxf(r.w, 0.f);
  }
  ((float4*)out)[i] = r;
}

// ---------------------------------------------------------------- launch
extern "C" void kernel_launch(void* const* d_in, const int* in_sizes, int n_in,
                              void* d_out, int out_size, void* d_ws, size_t ws_size,
                              hipStream_t stream) {
  const float* node_hidden = (const float*)d_in[0];   // [N,128]
  const float* edge_hidden = (const float*)d_in[1];   // [E,16]
  const int*   edge_index  = (const int*)  d_in[2];   // [2,E]
  const float* Wall        = (const float*)d_in[3];   // [3,128,128]
  const float* ball        = (const float*)d_in[4];   // [3,128]

  const int N = in_sizes[0] / DHID;
  const int E = in_sizes[2] / 2;
  const int L = in_sizes[3] / (DHID * DHID);

  const int* src = edge_index;
  const int* dst = edge_index + E;

  // workspace carve-up (all offsets 16B aligned: N*4, E*4, N*D*4 all %16==0)
  float* deg    = (float*)d_ws;                    // N
  float* invdeg = deg + N;                          // N
  float* ne     = invdeg + N;                       // E
  float* xbuf   = ne + E;                           // N*D
  float* ybuf   = xbuf + (size_t)N * DHID;          // N*D
  float* aggbuf = ybuf + (size_t)N * DHID;          // N*D

  const int ND   = N * DHID;
  const int ND4  = ND / 4;
  const int TPB  = 256;

  // --- gcn_norm precompute: deg = 1 + scatter(1), rsqrt, inv, per-edge norm
  k_fill1    <<<(N + TPB - 1) / TPB, TPB, 0, stream>>>(deg, N);
  k_count_deg<<<(E + TPB - 1) / TPB, TPB, 0, stream>>>(dst, deg, E);
  k_node_norm<<<(N + TPB - 1) / TPB, TPB, 0, stream>>>(deg, invdeg, N);
  k_edge_norm<<<(E + TPB - 1) / TPB, TPB, 0, stream>>>(src, dst, deg, ne, E);

  const int gemmGrid = (N + 127) / 128;
  const int scatGrid = (int)(((long long)E * 32 + TPB - 1) / TPB);

  const float* xin = node_hidden;
  for (int l = 0; l < L; ++l) {
    k_zero4   <<<(ND4 + TPB - 1) / TPB, TPB, 0, stream>>>((float4*)aggbuf, ND4);
    k_gemm    <<<gemmGrid, TPB, 0, stream>>>(xin, Wall + (size_t)l * DHID * DHID, ybuf, N);
    k_scatter <<<scatGrid, TPB, 0, stream>>>(ybuf, src, dst, ne, aggbuf, E);
    float* outp = (l == L - 1) ? (float*)d_out : xbuf;
    k_finalize<<<(ND4 + TPB - 1) / TPB, TPB, 0, stream>>>(
        aggbuf, ybuf, invdeg, ball + (size_t)l * DHID, outp, (l < L - 1) ? 1 : 0, ND4);
    xin = xbuf;
  }

  // pass-through: edge_hidden -> tail of d_out
  const int EH4 = (E * 16) / 4;
  k_copy4<<<(EH4 + TPB - 1) / TPB, TPB, 0, stream>>>(
      (const float4*)edge_hidden, (float4*)((float*)d_out + (size_t)N * DHID), EH4);
}